// GATLayer_36240934044288
// MI455X (gfx1250) — compile-verified
//
#include <hip/hip_runtime.h>
#include <math.h>

#define IN_DIM 256
#define OUT_DIM 64
#define NEG_SLOPE 0.01f
#define TILES_PER_BLOCK 4

typedef __attribute__((ext_vector_type(2))) float v2f;
typedef __attribute__((ext_vector_type(8))) float v8f;

// Order-preserving float <-> uint mapping so we can use integer atomicMax
// (lowers to GLOBAL_ATOMIC_MAX_U32) for the per-destination segment max.
__device__ __forceinline__ unsigned f2ord(float f) {
    unsigned u = __float_as_uint(f);
    return (u & 0x80000000u) ? ~u : (u | 0x80000000u);
}
__device__ __forceinline__ float ord2f(unsigned u) {
    return (u & 0x80000000u) ? __uint_as_float(u & 0x7FFFFFFFu)
                             : __uint_as_float(~u);
}

// ---------------------------------------------------------------------------
// Pass 0: zero h_out, init segment-max to -inf (ordered encoding), denom to 0.
// ---------------------------------------------------------------------------
__global__ void gat_init(float* __restrict__ h_out, unsigned* __restrict__ emax_u,
                         float* __restrict__ denom, int n_nodes) {
    int i = blockIdx.x * blockDim.x + threadIdx.x;
    if (i < n_nodes * OUT_DIM) h_out[i] = 0.0f;
    if (i < n_nodes) {
        emax_u[i] = 0x007FFFFFu;   // f2ord(-inf)
        denom[i]  = 0.0f;
    }
}

// ---------------------------------------------------------------------------
// Pass 1: z = h @ fc_w via V_WMMA_F32_16X16X4_F32 (fp32, matches reference).
// Block = 128 threads (4 waves). fc_w (64 KB) staged in LDS once per block.
// Each wave owns a 16x16 output tile (wave w -> columns w*16..w*16+15);
// block iterates over TILES_PER_BLOCK row tiles of 16 nodes.
// A-frag: lane l holds row (l&15), K pair {2*(l>>4), +1}  -> float2 load.
// B-frag: lane l holds col (l&15), same K pair            -> ds_load_2addr.
// D: lane l holds col (l&15), rows r + 8*(l>>4), r=0..7.
// Store epilogue: block-uniform full-tile fast path (no per-lane masking).
// ---------------------------------------------------------------------------
__global__ __launch_bounds__(128) void gat_fc_wmma(const float* __restrict__ h,
                                                   const float* __restrict__ fc_w,
                                                   float* __restrict__ z,
                                                   int n_nodes) {
    __shared__ float lds_w[IN_DIM * OUT_DIM];   // 64 KB
    const int tid = threadIdx.x;

    // Cooperative 128-bit fill of LDS with fc_w.
    const float4* s4 = (const float4*)fc_w;
    float4* d4 = (float4*)lds_w;
    for (int i = tid; i < (IN_DIM * OUT_DIM) / 4; i += 128) d4[i] = s4[i];
    __syncthreads();

    const int wave = tid >> 5;
    const int lane = tid & 31;
    const int mrow = lane & 15;          // A row within tile
    const int ncol = lane & 15;          // B/D column within tile
    const int kk   = (lane >> 4) << 1;   // K sub-offset: 0 or 2
    const int hi   = lane >> 4;          // D row-half selector
    const int n0   = wave * 16;

    const int tile_base = blockIdx.x * TILES_PER_BLOCK;
    for (int t = 0; t < TILES_PER_BLOCK; ++t) {
        const int m0 = (tile_base + t) * 16;
        if (m0 >= n_nodes) return;       // block-uniform exit
        const int mr = min(m0 + mrow, n_nodes - 1);
        const float* hrow = h + (size_t)mr * IN_DIM + kk;

        v8f acc = {};
#pragma unroll 4
        for (int k0 = 0; k0 < IN_DIM; k0 += 4) {
            v2f a = *(const v2f*)(hrow + k0);                       // contiguous K pair
            v2f b;
            b.x = lds_w[(k0 + kk) * OUT_DIM + n0 + ncol];
            b.y = lds_w[(k0 + kk + 1) * OUT_DIM + n0 + ncol];
            acc = __builtin_amdgcn_wmma_f32_16x16x4_f32(
                false, a, false, b, (short)0, acc, false, false);
        }

        float* zout = z + ((size_t)(m0 + 8 * hi)) * OUT_DIM + n0 + ncol;
        if (m0 + 16 <= n_nodes) {
            // Full tile: unguarded, EXEC stays all-ones.
#pragma unroll
            for (int r = 0; r < 8; ++r)
                zout[(size_t)r * OUT_DIM] = acc[r];
        } else {
            // Ragged tail tile (not hit for N % 16 == 0).
#pragma unroll
            for (int r = 0; r < 8; ++r)
                if (m0 + 8 * hi + r < n_nodes)
                    zout[(size_t)r * OUT_DIM] = acc[r];
        }
    }
}

// ---------------------------------------------------------------------------
// Pass 2: per-node attention scores s_src = z.a_l, s_dst = z.a_r.
// One wave per node: lane covers columns {lane, lane+32}, shuffle-reduce.
// ---------------------------------------------------------------------------
__global__ __launch_bounds__(256) void gat_scores(const float* __restrict__ z,
                                                  const float* __restrict__ attn_w,
                                                  float* __restrict__ s_src,
                                                  float* __restrict__ s_dst,
                                                  int n_nodes) {
    int gt = blockIdx.x * blockDim.x + threadIdx.x;
    int node = gt >> 5;
    int lane = threadIdx.x & 31;
    if (node >= n_nodes) return;
    const float* zr = z + (size_t)node * OUT_DIM;
    float z0 = zr[lane], z1 = zr[lane + 32];
    float ps = z0 * attn_w[lane]           + z1 * attn_w[lane + 32];
    float pd = z0 * attn_w[OUT_DIM + lane] + z1 * attn_w[OUT_DIM + lane + 32];
#pragma unroll
    for (int off = 16; off > 0; off >>= 1) {
        ps += __shfl_down(ps, off, 32);
        pd += __shfl_down(pd, off, 32);
    }
    if (lane == 0) { s_src[node] = ps; s_dst[node] = pd; }
}

// ---------------------------------------------------------------------------
// Pass 3: e = leaky_relu(s_src[src]+s_dst[dst]); segment max over dst.
// ---------------------------------------------------------------------------
__global__ void gat_edge_max(const int* __restrict__ src, const int* __restrict__ dst,
                             const float* __restrict__ s_src, const float* __restrict__ s_dst,
                             float* __restrict__ e_buf, unsigned* __restrict__ emax_u,
                             int n_edges) {
    int e = blockIdx.x * blockDim.x + threadIdx.x;
    if (e >= n_edges) return;
    int d = dst[e];
    float v = s_src[src[e]] + s_dst[d];
    v = v > 0.0f ? v : NEG_SLOPE * v;
    e_buf[e] = v;
    atomicMax(&emax_u[d], f2ord(v));
}

// ---------------------------------------------------------------------------
// Pass 4: ex = exp(e - emax[dst]) (in place); denom[dst] += ex.
// ---------------------------------------------------------------------------
__global__ void gat_edge_exp(const int* __restrict__ dst, const unsigned* __restrict__ emax_u,
                             float* __restrict__ e_buf, float* __restrict__ denom,
                             int n_edges) {
    int e = blockIdx.x * blockDim.x + threadIdx.x;
    if (e >= n_edges) return;
    int d = dst[e];
    float ex = __expf(e_buf[e] - ord2f(emax_u[d]));
    e_buf[e] = ex;
    atomicAdd(&denom[d], ex);
}

// ---------------------------------------------------------------------------
// Pass 5: weighted scatter h_out[dst] += alpha * z[src]. One wave per edge;
// lane covers columns {lane, lane+32} -> coalesced 32-lane f32 atomic bursts
// into L2-resident h_out (25.6 MB << 192 MB L2).
// ---------------------------------------------------------------------------
__global__ __launch_bounds__(256) void gat_scatter(const int* __restrict__ src,
                                                   const int* __restrict__ dst,
                                                   const float* __restrict__ e_buf,
                                                   const float* __restrict__ denom,
                                                   const float* __restrict__ z,
                                                   float* __restrict__ h_out,
                                                   int n_edges) {
    int gt = blockIdx.x * blockDim.x + threadIdx.x;
    int e = gt >> 5;
    int lane = threadIdx.x & 31;
    if (e >= n_edges) return;
    int s = src[e], d = dst[e];
    float alpha = e_buf[e] / fmaxf(denom[d], 1e-9f);
    const float* zr = z + (size_t)s * OUT_DIM;
    float* ho = h_out + (size_t)d * OUT_DIM;
    atomicAdd(&ho[lane],      alpha * zr[lane]);
    atomicAdd(&ho[lane + 32], alpha * zr[lane + 32]);
}

extern "C" void kernel_launch(void* const* d_in, const int* in_sizes, int n_in,
                              void* d_out, int out_size, void* d_ws, size_t ws_size,
                              hipStream_t stream) {
    const float* h      = (const float*)d_in[0];
    const float* fc_w   = (const float*)d_in[1];
    const float* attn_w = (const float*)d_in[2];
    const int*   src    = (const int*)d_in[3];
    const int*   dst    = (const int*)d_in[4];
    const int n_nodes = in_sizes[0] / IN_DIM;
    const int n_edges = in_sizes[3];
    float* h_out = (float*)d_out;

    // Workspace carve-up (~34 MB).
    float*    z      = (float*)d_ws;                          // N*64
    float*    e_buf  = z + (size_t)n_nodes * OUT_DIM;         // E (e, then exp in place)
    unsigned* emax_u = (unsigned*)(e_buf + n_edges);          // N
    float*    denom  = (float*)(emax_u + n_nodes);            // N
    float*    s_src  = denom + n_nodes;                       // N
    float*    s_dst  = s_src + n_nodes;                       // N

    const int initN = n_nodes * OUT_DIM;
    gat_init<<<(initN + 255) / 256, 256, 0, stream>>>(h_out, emax_u, denom, n_nodes);

    const int ntiles  = (n_nodes + 15) / 16;
    const int nblocks = (ntiles + TILES_PER_BLOCK - 1) / TILES_PER_BLOCK;
    gat_fc_wmma<<<nblocks, 128, 0, stream>>>(h, fc_w, z, n_nodes);

    long long score_threads = (long long)n_nodes * 32;
    gat_scores<<<(int)((score_threads + 255) / 256), 256, 0, stream>>>(z, attn_w, s_src, s_dst, n_nodes);

    gat_edge_max<<<(n_edges + 255) / 256, 256, 0, stream>>>(src, dst, s_src, s_dst, e_buf, emax_u, n_edges);
    gat_edge_exp<<<(n_edges + 255) / 256, 256, 0, stream>>>(dst, emax_u, e_buf, denom, n_edges);

    long long scat_threads = (long long)n_edges * 32;
    gat_scatter<<<(int)((scat_threads + 255) / 256), 256, 0, stream>>>(src, dst, e_buf, denom, z, h_out, n_edges);
}